// AdaptConv_78056735637935
// MI455X (gfx1250) — compile-verified
//
#include <hip/hip_runtime.h>
#include <math.h>

typedef float v2f __attribute__((ext_vector_type(2)));
typedef float v8f __attribute__((ext_vector_type(8)));

#define B_    8
#define N_    2048
#define C_    128
#define MT    64        // output rows per block (4 waves x 16 rows)
#define KC    64        // q-chunk size (4 x 16 subtiles, interleaved WMMA chains)
#define HS    132       // LDS row stride (floats) for h_q tile (conflict-free b64 B reads)
#define SS    20        // LDS row stride (floats) for gated-S tile (16B-aligned stores)
#define WAVES (MT/16)
#define NT    (KC/16)   // q-subtiles per chunk = independent GEMM1 chains

// ---------------- Kernel A: h = x @ W^T + b ----------------
__global__ __launch_bounds__(256) void linear_kernel(const float* __restrict__ x,
                                                     const float* __restrict__ W,
                                                     const float* __restrict__ bias,
                                                     float* __restrict__ h) {
  constexpr int ROWS = 32;
  __shared__ __align__(16) float xs[ROWS * C_];
  const int tid  = threadIdx.x;
  const int row0 = blockIdx.x * ROWS;

  const float4* xg = (const float4*)(x + (size_t)row0 * C_);
  for (int i = tid; i < ROWS * C_ / 4; i += 256)
    ((float4*)xs)[i] = xg[i];
  __syncthreads();

  const int c    = tid & 127;
  const int half = tid >> 7;

  float4 w[32];
  const float4* W4 = (const float4*)W;
#pragma unroll
  for (int i = 0; i < 32; ++i) w[i] = W4[c * 32 + i];
  const float bc = bias[c];

  for (int r = half; r < ROWS; r += 2) {
    float acc = bc;
    const float4* xr = (const float4*)&xs[r * C_];
#pragma unroll
    for (int i = 0; i < 32; ++i) {
      float4 xv = xr[i];
      acc += w[i].x * xv.x + w[i].y * xv.y + w[i].z * xv.z + w[i].w * xv.w;
    }
    h[(size_t)(row0 + r) * C_ + c] = acc;
  }
}

// ---------------- Kernel B: rnorm = 1 / max(||h_row||, eps) ----------------
__global__ __launch_bounds__(32) void rnorm_kernel(const float* __restrict__ h,
                                                   float* __restrict__ rn) {
  const int row  = blockIdx.x;
  const int lane = threadIdx.x;
  const float4* h4 = (const float4*)(h + (size_t)row * C_);
  float4 v = h4[lane];
  float s = v.x * v.x + v.y * v.y + v.z * v.z + v.w * v.w;
#pragma unroll
  for (int off = 16; off > 0; off >>= 1) s += __shfl_xor(s, off, 32);
  if (lane == 0) rn[row] = 1.0f / fmaxf(sqrtf(s), 1e-12f);
}

// ---------------- Kernel C: fused cos / gate / aggregate / relu ----------------
__global__ __launch_bounds__(WAVES * 32) void fused_kernel(const float* __restrict__ ew,
                                                           const float* __restrict__ h,
                                                           const float* __restrict__ rn,
                                                           float* __restrict__ out) {
  __shared__ __align__(16) float hq[KC * HS];           // shared h_q tile (~33 KB)
  __shared__ __align__(16) float sred[WAVES * KC * SS]; // per-wave gated-S tile, [q][m] (~20 KB)

  const int tid  = threadIdx.x;
  const int wave = tid >> 5;
  const int lane = tid & 31;
  const int nl   = lane & 15;   // N index within 16-tile
  const int hi   = lane >> 4;   // half selector

  const int b     = blockIdx.x / (N_ / MT);
  const int p0    = (blockIdx.x % (N_ / MT)) * MT;
  const int prow0 = p0 + wave * 16;

  const float* hb  = h  + (size_t)b * N_ * C_;
  const float* rb  = rn + (size_t)b * N_;
  const float* ewb = ew + (size_t)b * N_ * N_;

  // A-fragments of this wave's 16x128 h_p strip: lane(m=nl) holds K = 4kk+2hi, +1
  v2f a1[32];
  {
    const float* hp = hb + (size_t)(prow0 + nl) * C_ + 2 * hi;
#pragma unroll
    for (int kk = 0; kk < 32; ++kk) a1[kk] = *(const v2f*)(hp + kk * 4);
  }
  float rp[8];
#pragma unroll
  for (int r = 0; r < 8; ++r) rp[r] = rb[prow0 + 8 * hi + r];

  const v8f vzero = {0, 0, 0, 0, 0, 0, 0, 0};
  v8f acc[8];
#pragma unroll
  for (int ct = 0; ct < 8; ++ct) acc[ct] = vzero;

  for (int q0 = 0; q0 < N_; q0 += KC) {
    __syncthreads();  // protect hq from previous iteration's readers
    const float4* src = (const float4*)(hb + (size_t)q0 * C_);
    for (int i = tid; i < KC * C_ / 4; i += WAVES * 32) {
      int row = i >> 5, c4 = i & 31;
      *(float4*)&hq[row * HS + c4 * 4] = src[i];
    }
    __syncthreads();

    float* sw = &sred[wave * KC * SS];

    // ---- GEMM1: NT independent S chains, interleaved over shared A fragments ----
    v8f s[NT];
#pragma unroll
    for (int t = 0; t < NT; ++t) s[t] = vzero;
    const float* brow = &hq[nl * HS + 2 * hi];
#pragma unroll
    for (int kk = 0; kk < 32; ++kk) {
#pragma unroll
      for (int t = 0; t < NT; ++t) {
        v2f bf = *(const v2f*)(brow + t * 16 * HS + kk * 4);
        s[t] = __builtin_amdgcn_wmma_f32_16x16x4_f32(false, a1[kk], false, bf,
                                                     (short)0, s[t], false, false);
      }
    }

    // ---- gating + spill to LDS in A-layout [q][m] ----
#pragma unroll
    for (int t = 0; t < NT; ++t) {
      const int qb = q0 + t * 16;
      const float rq = rb[qb + nl];
      const float* ewp = ewb + (size_t)(prow0 + 8 * hi) * N_ + qb + nl;
      float g[8];
#pragma unroll
      for (int r = 0; r < 8; ++r)
        g[r] = s[t][r] * rp[r] * rq * ewp[(size_t)r * N_];
      float* sp = &sw[(t * 16 + nl) * SS + 8 * hi];
      *(float4*)sp       = make_float4(g[0], g[1], g[2], g[3]);
      *(float4*)(sp + 4) = make_float4(g[4], g[5], g[6], g[7]);
    }

    // ---- GEMM2: acc += gated(16xKC) @ h_q(KCx128), 8 independent chains ----
#pragma unroll
    for (int kk2 = 0; kk2 < KC / 4; ++kk2) {
      const int k0 = kk2 * 4 + 2 * hi;
      v2f a2;
      a2.x = sw[k0 * SS + nl];
      a2.y = sw[(k0 + 1) * SS + nl];
#pragma unroll
      for (int ct = 0; ct < 8; ++ct) {
        const float* hp2 = &hq[k0 * HS + ct * 16 + nl];
        v2f b2;
        b2.x = hp2[0];
        b2.y = hp2[HS];
        acc[ct] = __builtin_amdgcn_wmma_f32_16x16x4_f32(false, a2, false, b2,
                                                        (short)0, acc[ct], false, false);
      }
    }
  }

  // ---- Epilogue: ReLU + store ----
  float* ob = out + (size_t)b * N_ * C_;
#pragma unroll
  for (int ct = 0; ct < 8; ++ct)
#pragma unroll
    for (int r = 0; r < 8; ++r) {
      float v = fmaxf(acc[ct][r], 0.0f);
      ob[(size_t)(prow0 + 8 * hi + r) * C_ + ct * 16 + nl] = v;
    }
}

extern "C" void kernel_launch(void* const* d_in, const int* in_sizes, int n_in,
                              void* d_out, int out_size, void* d_ws, size_t ws_size,
                              hipStream_t stream) {
  (void)in_sizes; (void)n_in; (void)out_size; (void)ws_size;
  const float* x    = (const float*)d_in[0];
  const float* ew   = (const float*)d_in[1];
  const float* W    = (const float*)d_in[2];
  const float* bias = (const float*)d_in[3];
  float* out = (float*)d_out;

  float* h  = (float*)d_ws;                       // 8*2048*128 floats = 8 MB
  float* rn = h + (size_t)B_ * N_ * C_;           // 8*2048 floats

  linear_kernel<<<B_ * N_ / 32, 256, 0, stream>>>(x, W, bias, h);
  rnorm_kernel<<<B_ * N_, 32, 0, stream>>>(h, rn);
  fused_kernel<<<B_ * (N_ / MT), WAVES * 32, 0, stream>>>(ew, h, rn, out);
}